// CenterlineLoss_2714419331840
// MI455X (gfx1250) — compile-verified
//
#include <hip/hip_runtime.h>
#include <hip/hip_bf16.h>

typedef __attribute__((ext_vector_type(2))) float v2f;
typedef __attribute__((ext_vector_type(8))) float v8f;

#define BOUNDV 2000.0f
#define INF_F  __builtin_huge_valf()

// ---------------------------------------------------------------------------
// Kernel A (min1 direction): one wave per 16-row bez tile, loops ref tiles.
// Tracks  rmin[i] = min_n (normR_n - 2*dot[m][n])  (2 VALU per i per tile);
// row constant (normB) added once at the end; masked rows are skipped in the sum. No atomics.
// ---------------------------------------------------------------------------
__global__ void cl_min1_wmma_kernel(const float* __restrict__ bez,
                                    const float* __restrict__ ref,
                                    float* __restrict__ sum1P,
                                    unsigned int* __restrict__ cntP,
                                    int nB, int nR) {
    const int gtid = blockIdx.x * blockDim.x + threadIdx.x;
    const int wave = gtid >> 5;
    const int lane = gtid & 31;
    const int half = lane >> 4;
    const int l16  = lane & 15;

    const int mBase = wave * 16;
    if (mBase >= nB) return;               // wave-uniform

    // A fragment: rows = bez points. Lanes 0-15: (bx,by); lanes 16-31: K pad.
    const int mIdx = mBase + l16;
    const float ax = bez[2 * mIdx + 0];
    const float ay = bez[2 * mIdx + 1];
    const bool  mOK = (ax >= -BOUNDV) & (ax <= BOUNDV) &
                      (ay >= -BOUNDV) & (ay <= BOUNDV);
    const float normB  = ax * ax + ay * ay;
    const float maskf  = mOK ? 1.0f : 0.0f;

    v2f a;
    a.x = half ? 0.0f : ax;
    a.y = half ? 0.0f : ay;

    // hoisted row broadcasts (loop-invariant)
    float baseRow[8], mkRow[8];
#pragma unroll
    for (int i = 0; i < 8; ++i) {
        baseRow[i] = __shfl(normB, i + 8 * half, 32);
        mkRow[i]   = __shfl(maskf, i + 8 * half, 32);
    }

    float rmin[8];
#pragma unroll
    for (int i = 0; i < 8; ++i) rmin[i] = INF_F;

    const v2f* ref2 = (const v2f*)ref;     // (mem order: [y-coord, x-coord] after swap)
    const int nTiles = nR >> 4;

    v2f rcur = ref2[l16];                  // prefetch tile 0
#pragma unroll 4
    for (int rt = 0; rt < nTiles; ++rt) {
        const v2f r = rcur;
        const int rtn = (rt + 1 < nTiles) ? (rt + 1) : rt;
        rcur = ref2[rtn * 16 + l16];       // prefetch next tile

        // ref columns swapped per reference: point = (mem[1], mem[0])
        const float rx = r.y;
        const float ry = r.x;
        const float normR = rx * rx + ry * ry;

        v2f b;
        b.x = half ? 0.0f : rx;
        b.y = half ? 0.0f : ry;

        v8f c = {};
        c = __builtin_amdgcn_wmma_f32_16x16x4_f32(
                false, a, false, b, (short)0, c, false, false);

#pragma unroll
        for (int i = 0; i < 8; ++i) {
            const float t = __builtin_fmaf(c[i], -2.0f, normR);
            rmin[i] = fminf(rmin[i], t);
        }
    }

    // min over all n: butterfly within each 16-lane half
#pragma unroll
    for (int i = 0; i < 8; ++i) {
        float v = rmin[i];
        v = fminf(v, __shfl_xor(v, 1, 32));
        v = fminf(v, __shfl_xor(v, 2, 32));
        v = fminf(v, __shfl_xor(v, 4, 32));
        v = fminf(v, __shfl_xor(v, 8, 32));
        rmin[i] = v;
    }

    // masked sum of per-row distances
    float s = 0.0f;
#pragma unroll
    for (int i = 0; i < 8; ++i) {
        const float d2 = rmin[i] + baseRow[i];
        const float dist = sqrtf(fmaxf(d2, 0.0f));
        if (mkRow[i] != 0.0f) s += dist;
    }
    s += __shfl_xor(s, 16, 32);

    unsigned long long bal = __ballot(mOK && half == 0);
    if (lane == 0) {
        sum1P[wave] = s;
        cntP[wave]  = (unsigned int)__popcll(bal);
    }
}

// ---------------------------------------------------------------------------
// Kernel B (min2 direction): mirror image. One wave per 16-row REF tile,
// loops bez tiles. bez bounds-mask folds in per-lane as +inf (no shuffles
// in the loop). Each wave owns its ref rows' minima -> direct stores,
// zero atomics, no init pass needed.
// ---------------------------------------------------------------------------
__global__ void cl_min2_wmma_kernel(const float* __restrict__ bez,
                                    const float* __restrict__ ref,
                                    float* __restrict__ min2sq,
                                    int nB, int nR) {
    const int gtid = blockIdx.x * blockDim.x + threadIdx.x;
    const int wave = gtid >> 5;
    const int lane = gtid & 31;
    const int half = lane >> 4;
    const int l16  = lane & 15;

    const int rBase = wave * 16;
    if (rBase >= nR) return;               // wave-uniform

    // A fragment: rows = ref points (columns swapped)
    const int rIdx = rBase + l16;
    const float rx = ref[2 * rIdx + 1];
    const float ry = ref[2 * rIdx + 0];
    const float normR = rx * rx + ry * ry;

    v2f a;
    a.x = half ? 0.0f : rx;
    a.y = half ? 0.0f : ry;

    // hoisted row constants (added after the reduction)
    float normRRow[8];
#pragma unroll
    for (int i = 0; i < 8; ++i)
        normRRow[i] = __shfl(normR, i + 8 * half, 32);

    float rmin[8];
#pragma unroll
    for (int i = 0; i < 8; ++i) rmin[i] = INF_F;

    const v2f* bez2 = (const v2f*)bez;
    const int nTiles = nB >> 4;

    v2f bcur = bez2[l16];                  // prefetch tile 0
#pragma unroll 4
    for (int bt = 0; bt < nTiles; ++bt) {
        const v2f p = bcur;
        const int btn = (bt + 1 < nTiles) ? (bt + 1) : bt;
        bcur = bez2[btn * 16 + l16];       // prefetch next tile

        const float bx = p.x;
        const float by = p.y;
        const bool ok = (bx >= -BOUNDV) & (bx <= BOUNDV) &
                        (by >= -BOUNDV) & (by <= BOUNDV);
        // masked bez column -> +inf, so it never wins the min
        const float normBm = ok ? (bx * bx + by * by) : INF_F;

        v2f b;
        b.x = half ? 0.0f : bx;
        b.y = half ? 0.0f : by;

        v8f c = {};
        // dot[refRow][bezCol], refRow = i + 8*half, bezCol = l16
        c = __builtin_amdgcn_wmma_f32_16x16x4_f32(
                false, a, false, b, (short)0, c, false, false);

#pragma unroll
        for (int i = 0; i < 8; ++i) {
            const float t = __builtin_fmaf(c[i], -2.0f, normBm);
            rmin[i] = fminf(rmin[i], t);
        }
    }

    // min over all bez: butterfly within each 16-lane half
#pragma unroll
    for (int i = 0; i < 8; ++i) {
        float v = rmin[i];
        v = fminf(v, __shfl_xor(v, 1, 32));
        v = fminf(v, __shfl_xor(v, 2, 32));
        v = fminf(v, __shfl_xor(v, 4, 32));
        v = fminf(v, __shfl_xor(v, 8, 32));
        rmin[i] = v;
    }

    // lanes 0 and 16 store their half's 8 row minima (squared)
    if (l16 == 0) {
#pragma unroll
        for (int i = 0; i < 8; ++i)
            min2sq[rBase + 8 * half + i] = rmin[i] + normRRow[i];
    }
}

// ---------------------------------------------------------------------------
// finalize: fixed-order single-block reduction -> scalar output.
// ---------------------------------------------------------------------------
__global__ void cl_finalize_kernel(const float* __restrict__ min2sq,
                                   const float* __restrict__ sum1P,
                                   const unsigned int* __restrict__ cntP,
                                   float* __restrict__ out,
                                   int nR, int nWaves) {
    __shared__ float sm[256];
    const int t = threadIdx.x;

    float s2 = 0.0f;
    for (int j = t; j < nR; j += 256)
        s2 += sqrtf(fmaxf(min2sq[j], 0.0f));
    sm[t] = s2;
    __syncthreads();
    for (int o = 128; o > 0; o >>= 1) {
        if (t < o) sm[t] += sm[t + o];
        __syncthreads();
    }
    const float mean2 = sm[0] / (float)nR;
    __syncthreads();

    float s1 = 0.0f;
    for (int j = t; j < nWaves; j += 256) s1 += sum1P[j];
    sm[t] = s1;
    __syncthreads();
    for (int o = 128; o > 0; o >>= 1) {
        if (t < o) sm[t] += sm[t + o];
        __syncthreads();
    }
    const float sum1 = sm[0];
    __syncthreads();

    float cf = 0.0f;
    for (int j = t; j < nWaves; j += 256) cf += (float)cntP[j];
    sm[t] = cf;
    __syncthreads();
    for (int o = 128; o > 0; o >>= 1) {
        if (t < o) sm[t] += sm[t + o];
        __syncthreads();
    }

    if (t == 0) {
        const float nValid = sm[0];
        const float mean1 = sum1 / nValid;
        out[0] = 0.5f * (mean1 + mean2);
    }
}

// ---------------------------------------------------------------------------
extern "C" void kernel_launch(void* const* d_in, const int* in_sizes, int n_in,
                              void* d_out, int out_size, void* d_ws, size_t ws_size,
                              hipStream_t stream) {
    const float* bez = (const float*)d_in[0];   // (nB, 2) f32
    const float* ref = (const float*)d_in[1];   // (nR, 2) f32
    float* out = (float*)d_out;

    const int nB = in_sizes[0] / 2;   // 16384
    const int nR = in_sizes[1] / 2;   // 8192
    const int nWavesA = nB / 16;      // 1024
    const int nWavesB = nR / 16;      // 512

    // workspace: [ min2sq: nR f32 | sum1P: nWavesA f32 | cntP: nWavesA u32 ]
    float*        min2sq = (float*)d_ws;
    float*        sum1P  = (float*)((char*)d_ws + (size_t)nR * 4);
    unsigned int* cntP   = (unsigned int*)((char*)d_ws + (size_t)nR * 4 +
                                           (size_t)nWavesA * 4);

    const int blkA = (nWavesA * 32 + 255) / 256;
    cl_min1_wmma_kernel<<<blkA, 256, 0, stream>>>(bez, ref, sum1P, cntP, nB, nR);

    const int blkB = (nWavesB * 32 + 255) / 256;
    cl_min2_wmma_kernel<<<blkB, 256, 0, stream>>>(bez, ref, min2sq, nB, nR);

    cl_finalize_kernel<<<1, 256, 0, stream>>>(min2sq, sum1P, cntP, out, nR, nWavesA);
}